// GraphRegressorEmbed_56298431316164
// MI455X (gfx1250) — compile-verified
//
#include <hip/hip_runtime.h>
#include <hip/hip_bf16.h>

#define N_NODES   100000
#define N_EDGES   1600000
#define NUM_GRAPHS 64
#define EMBED     32
#define LAYERS    9

typedef __attribute__((ext_vector_type(2))) float v2f;
typedef __attribute__((ext_vector_type(8))) float v8f;

__device__ __forceinline__ void atomAddF(float* p, float v) {
    // lowers to global_atomic_add_f32 (no CAS loop) on gfx1250
    unsafeAtomicAdd(p, v);
}

// ---------------- degree / norm ----------------
__global__ void k_init_deg(float* __restrict__ deg) {
    int n = blockIdx.x * blockDim.x + threadIdx.x;
    if (n < N_NODES) deg[n] = 1.0f;  // self loop
}

__global__ void k_deg_accum(const int* __restrict__ col, float* __restrict__ deg) {
    int e = blockIdx.x * blockDim.x + threadIdx.x;
    if (e < N_EDGES) atomAddF(&deg[col[e]], 1.0f);
}

__global__ void k_dinv(float* __restrict__ deg) {
    int n = blockIdx.x * blockDim.x + threadIdx.x;
    if (n < N_NODES) deg[n] = rsqrtf(deg[n]);
}

// ---------------- embed: h = x @ embed_W + embed_b ----------------
__global__ void k_embed(const float* __restrict__ x, const float* __restrict__ W,
                        const float* __restrict__ b, float* __restrict__ h) {
    int idx = blockIdx.x * blockDim.x + threadIdx.x;
    if (idx < N_NODES * EMBED) {
        int n = idx >> 5, c = idx & 31;
        h[idx] = x[2 * n] * W[c] + x[2 * n + 1] * W[EMBED + c] + b[c];
    }
}

// ---------------- per-layer linear via fp32 WMMA ----------------
// t = relu?(hin) @ W ; agg = bias + dinv[n]^2 * t[n]   (self-loop term fused)
// One wave: 16 nodes x 32 cols = two 16x16 C tiles, K=32 as 8 steps of K=4.
// APPLY_RELU is a template parameter so the relu=false path has zero select
// overhead and the relu=true path is a bare v_max_num_f32 on the A fragment.
template <bool APPLY_RELU>
__global__ __launch_bounds__(256)
void k_gcn_lin_wmma(const float* __restrict__ hin,
                    const float* __restrict__ W,     // [32,32] this layer
                    const float* __restrict__ bias,  // [32]
                    const float* __restrict__ dinv,
                    float* __restrict__ t,
                    float* __restrict__ agg) {
    const int lane  = threadIdx.x & 31;
    const int wave  = threadIdx.x >> 5;
    const int node0 = (blockIdx.x * 8 + wave) * 16;
    if (node0 >= N_NODES) return;          // wave-uniform exit, EXEC stays full
    const int half = lane >> 4;            // 0: K pair {0,1}; 1: K pair {2,3}
    const int m16  = lane & 15;

    v8f acc0 = {};  // N tile 0..15
    v8f acc1 = {};  // N tile 16..31
#pragma unroll
    for (int k0 = 0; k0 < EMBED; k0 += 4) {
        // A fragment: lane holds A[m16][k0+2*half], A[m16][k0+2*half+1]
        v2f a = *(const v2f*)(hin + (node0 + m16) * EMBED + k0 + 2 * half);
        if (APPLY_RELU) { a.x = fmaxf(a.x, 0.0f); a.y = fmaxf(a.y, 0.0f); }
        // B fragments: lane holds B[k0+2*half][col], B[k0+2*half+1][col]
        const float* bp = W + (k0 + 2 * half) * EMBED + m16;
        v2f b0, b1;
        b0.x = bp[0];          b0.y = bp[EMBED];
        b1.x = bp[16];         b1.y = bp[EMBED + 16];
        acc0 = __builtin_amdgcn_wmma_f32_16x16x4_f32(false, a, false, b0,
                                                     (short)0, acc0, false, false);
        acc1 = __builtin_amdgcn_wmma_f32_16x16x4_f32(false, a, false, b1,
                                                     (short)0, acc1, false, false);
    }
    // C/D layout: VGPR r -> row (r + 8*half), col = m16 (+16 for tile 1)
#pragma unroll
    for (int r = 0; r < 8; ++r) {
        int   row   = node0 + r + 8 * half;
        float dn    = dinv[row];
        float selfn = dn * dn;
        float v0 = acc0[r], v1 = acc1[r];
        t[row * EMBED + m16]        = v0;
        t[row * EMBED + m16 + 16]   = v1;
        agg[row * EMBED + m16]      = bias[m16]      + selfn * v0;
        agg[row * EMBED + m16 + 16] = bias[m16 + 16] + selfn * v1;
    }
}

// ---------------- edge scatter: agg[col] += dinv[row]*dinv[col]*t[row] ----------------
// 32 consecutive lanes cover one edge's 32 channels: coalesced gather + atomics.
__global__ __launch_bounds__(256)
void k_edge_scatter(const int* __restrict__ rowi, const int* __restrict__ coli,
                    const float* __restrict__ dinv,
                    const float* __restrict__ t, float* __restrict__ agg) {
    long long idx = (long long)blockIdx.x * blockDim.x + threadIdx.x;
    if (idx >= (long long)N_EDGES * EMBED) return;
    int e = (int)(idx >> 5);
    int c = (int)(idx & 31);
    int r  = rowi[e];
    int cl = coli[e];
    float nrm = dinv[r] * dinv[cl];
    atomAddF(&agg[cl * EMBED + c], nrm * t[r * EMBED + c]);
}

// ---------------- pooling + head ----------------
__global__ void k_pool_zero(float* __restrict__ pooled, float* __restrict__ counts) {
    int i = blockIdx.x * blockDim.x + threadIdx.x;
    if (i < NUM_GRAPHS * EMBED) pooled[i] = 0.0f;
    if (i < NUM_GRAPHS) counts[i] = 0.0f;
}

__global__ __launch_bounds__(256)
void k_pool_accum(const float* __restrict__ h, const int* __restrict__ batch,
                  float* __restrict__ pooled, float* __restrict__ counts) {
    int idx = blockIdx.x * blockDim.x + threadIdx.x;
    if (idx < N_NODES * EMBED) {
        int n = idx >> 5, c = idx & 31;
        int g = batch[n];
        atomAddF(&pooled[g * EMBED + c], fmaxf(h[idx], 0.0f));  // final relu fused
        if (c == 0) atomAddF(&counts[g], 1.0f);
    }
}

__global__ void k_final(const float* __restrict__ pooled, const float* __restrict__ counts,
                        const float* __restrict__ fcW, const float* __restrict__ fcb,
                        float* __restrict__ out) {
    int i = threadIdx.x;
    if (i < NUM_GRAPHS * 2) {
        int g = i >> 1, k = i & 1;
        float inv = 1.0f / fmaxf(counts[g], 1.0f);
        float s = 0.0f;
#pragma unroll
        for (int c = 0; c < EMBED; ++c) s += pooled[g * EMBED + c] * fcW[c * 2 + k];
        out[i] = s * inv + fcb[k];
    }
}

static inline size_t alignUp(size_t v, size_t a) { return (v + a - 1) & ~(a - 1); }

extern "C" void kernel_launch(void* const* d_in, const int* in_sizes, int n_in,
                              void* d_out, int out_size, void* d_ws, size_t ws_size,
                              hipStream_t stream) {
    (void)in_sizes; (void)n_in; (void)out_size; (void)ws_size;
    const float* x       = (const float*)d_in[0];
    const int*   eidx    = (const int*)d_in[1];     // [2, E] flat
    const int*   batch   = (const int*)d_in[2];
    const float* embed_W = (const float*)d_in[3];
    const float* embed_b = (const float*)d_in[4];
    const float* conv_W  = (const float*)d_in[5];   // [L,32,32]
    const float* conv_b  = (const float*)d_in[6];   // [L,32]
    const float* fc_W    = (const float*)d_in[7];   // [32,2]
    const float* fc_b    = (const float*)d_in[8];   // [2]
    float* out = (float*)d_out;

    const int* rowi = eidx;
    const int* coli = eidx + N_EDGES;

    // workspace layout (all L2-resident: ~38.8 MB)
    char*  ws   = (char*)d_ws;
    size_t off  = 0;
    float* dinv = (float*)(ws + off); off = alignUp(off + (size_t)N_NODES * 4, 256);
    float* hA   = (float*)(ws + off); off = alignUp(off + (size_t)N_NODES * EMBED * 4, 256);
    float* hB   = (float*)(ws + off); off = alignUp(off + (size_t)N_NODES * EMBED * 4, 256);
    float* tbuf = (float*)(ws + off); off = alignUp(off + (size_t)N_NODES * EMBED * 4, 256);
    float* pooled = (float*)(ws + off); off = alignUp(off + (size_t)NUM_GRAPHS * EMBED * 4, 256);
    float* counts = (float*)(ws + off); off = alignUp(off + (size_t)NUM_GRAPHS * 4, 256);

    const int B = 256;

    // degrees + symmetric norm factors
    k_init_deg<<<(N_NODES + B - 1) / B, B, 0, stream>>>(dinv);
    k_deg_accum<<<(N_EDGES + B - 1) / B, B, 0, stream>>>(coli, dinv);
    k_dinv<<<(N_NODES + B - 1) / B, B, 0, stream>>>(dinv);

    // embed
    k_embed<<<(N_NODES * EMBED + B - 1) / B, B, 0, stream>>>(x, embed_W, embed_b, hA);

    // 9 GCN layers, ping-pong hA/hB
    float* bufs[2] = { hA, hB };
    const int wmmaGrid = (N_NODES + 16 * 8 - 1) / (16 * 8);
    const long long edgeThreads = (long long)N_EDGES * EMBED;
    const int edgeGrid = (int)((edgeThreads + B - 1) / B);
    for (int l = 0; l < LAYERS; ++l) {
        float* hin = bufs[l & 1];
        float* agg = bufs[(l + 1) & 1];
        const float* Wl = conv_W + (size_t)l * EMBED * EMBED;
        const float* bl = conv_b + (size_t)l * EMBED;
        if (l == 0) {
            k_gcn_lin_wmma<false><<<wmmaGrid, B, 0, stream>>>(hin, Wl, bl, dinv, tbuf, agg);
        } else {
            k_gcn_lin_wmma<true><<<wmmaGrid, B, 0, stream>>>(hin, Wl, bl, dinv, tbuf, agg);
        }
        k_edge_scatter<<<edgeGrid, B, 0, stream>>>(rowi, coli, dinv, tbuf, agg);
    }
    float* hfinal = bufs[LAYERS & 1];

    // mean pool + head
    k_pool_zero<<<(NUM_GRAPHS * EMBED + B - 1) / B, B, 0, stream>>>(pooled, counts);
    k_pool_accum<<<(N_NODES * EMBED + B - 1) / B, B, 0, stream>>>(hfinal, batch, pooled, counts);
    k_final<<<1, 128, 0, stream>>>(pooled, counts, fc_W, fc_b, out);
}